// SimplifiedQMixer_16578573763435
// MI455X (gfx1250) — compile-verified
//
#include <hip/hip_runtime.h>
#include <math.h>

// MI455X / gfx1250 fused QMixer forward.
// One kernel: features -> hypernet GEMMs (WMMA f32 16x16x4) -> fused
// |h1@w1_2 + b1_2| * qs contraction -> elu -> weighted sum -> q_tot.

typedef __attribute__((ext_vector_type(2))) float v2f;
typedef __attribute__((ext_vector_type(8))) float v8f;

#define BTOT 32768   // B*T
#define NA   64      // N agents
#define OO   32      // obs dim
#define EE   32      // embed
#define HH   64      // hidden
#define SS   129     // state features
#define SP   132     // padded K (multiple of 4)
#define TT   256     // T

__device__ inline v8f wmma4(v2f a, v2f b, v8f c) {
  // D = A(16x4 f32) x B(4x16 f32) + C(16x16 f32)
  return __builtin_amdgcn_wmma_f32_16x16x4_f32(false, a, false, b, (short)0, c,
                                               false, false);
}

// 16x16 output tile of  relu-less  (s(16xSP) @ W(SS x ldW))[:, c0:c0+16] + bias
// s comes from per-wave LDS (zero-padded rows 129..131); W rows >= SS read as 0.
__device__ inline v8f gemm_s(const float* __restrict__ sW,
                             const float* __restrict__ W, int ldW,
                             const float* __restrict__ bias, int c0,
                             int ln15, int kb) {
  float bv = bias[c0 + ln15];
  v8f acc;
#pragma unroll
  for (int v = 0; v < 8; ++v) acc[v] = bv;
#pragma unroll
  for (int k = 0; k < SP; k += 4) {
    v2f a;
    a[0] = sW[ln15 * SP + k + kb];
    a[1] = sW[ln15 * SP + k + kb + 1];
    int kk = k + kb;
    v2f b;
    b[0] = (kk < SS) ? W[kk * ldW + c0 + ln15] : 0.f;
    b[1] = (kk + 1 < SS) ? W[(kk + 1) * ldW + c0 + ln15] : 0.f;
    acc = wmma4(a, b, acc);
  }
  return acc;
}

extern "C" __global__ __launch_bounds__(128)
void qmix_fused_kernel(const float* __restrict__ agent_qs,
                       const float* __restrict__ obs,
                       const int*   __restrict__ agent_mask,
                       const float* __restrict__ w1_1, const float* __restrict__ b1_1,
                       const float* __restrict__ w1_2, const float* __restrict__ b1_2,
                       const float* __restrict__ w2_1, const float* __restrict__ b2_1,
                       const float* __restrict__ w2_2, const float* __restrict__ b2_2,
                       const float* __restrict__ wb1,  const float* __restrict__ bb1,
                       const float* __restrict__ wb2_1,const float* __restrict__ bb2_1,
                       const float* __restrict__ wb2_2,const float* __restrict__ bb2_2,
                       float* __restrict__ out) {
  __shared__ float lds_s[4 * 16 * SP];   // per-wave state tiles (zero-padded K)
  __shared__ float lds_h[4 * 16 * HH];   // per-wave hidden tiles (h2 then h1)
  __shared__ float lds_w[HH * EE];       // shared w1_2 chunk for one n (64x32)

  const int tid  = threadIdx.x;
  const int lane = tid & 31;
  const int wid  = tid >> 5;
  const int ln15 = lane & 15;
  const int hi   = lane >> 4;   // 0 or 1
  const int kb   = hi << 1;     // K sub-pair for A/B fragments

  const int grow0 = blockIdx.x * 64 + wid * 16;  // first (b*T+t) row of wave
  const int bidx  = (blockIdx.x * 64) / TT;      // batch index (uniform in WG)

  float* sW = lds_s + wid * (16 * SP);
  float* hW = lds_h + wid * (16 * HH);

  // ---- active agent count k = sum(agent_mask[b,:]) ----
  int mv = agent_mask[bidx * NA + lane] + agent_mask[bidx * NA + 32 + lane];
#pragma unroll
  for (int off = 16; off >= 1; off >>= 1) mv += __shfl_xor(mv, off, 32);
  const int   cnt = mv;
  const float fc  = (float)cnt;

  // ---- Phase A: state features (lane <-> obs feature o) ----
  for (int r = 0; r < 16; ++r) {
    size_t base = ((size_t)(grow0 + r)) * (NA * OO) + lane;
    float sum = 0.f, ssq = 0.f, mn = INFINITY, mx = -INFINITY;
    for (int n = 0; n < cnt; ++n) {              // first-k agents are "active"
      float x = obs[base + (size_t)n * OO];
      sum += x; ssq += x * x;
      mn = fminf(mn, x); mx = fmaxf(mx, x);
    }
    float mean = (cnt > 0) ? (sum / fc) : 0.f;
    float varn = ssq - fc * mean * mean;          // sum((x-mean)^2)
    float stdv = (cnt > 1) ? sqrtf(fmaxf(varn, 0.f) / (fc - 1.f)) : 0.f;
    if (cnt == 0) { mn = 0.f; mx = 0.f; }
    sW[r * SP +       lane] = mean;
    sW[r * SP +  32 + lane] = stdv;
    sW[r * SP +  64 + lane] = mn;
    sW[r * SP +  96 + lane] = mx;
    if (lane < 4)  // ratio at 128, zero pad 129..131
      sW[r * SP + 128 + lane] = (lane == 0 && cnt > 0) ? fc * (1.f / 64.f) : 0.f;
  }

  // ---- Phase B: small hypernet GEMMs (per-wave, WMMA) ----
  // h2 = relu(s @ w2_1 + b2_1) -> lds_h
  for (int nt = 0; nt < 4; ++nt) {
    v8f acc = gemm_s(sW, w2_1, HH, b2_1, nt * 16, ln15, kb);
#pragma unroll
    for (int v = 0; v < 8; ++v)
      hW[(v + 8 * hi) * HH + nt * 16 + ln15] = fmaxf(acc[v], 0.f);
  }
  // w2 = |h2 @ w2_2 + b2_2|  (kept in fragment registers)
  v8f w2f[2];
  for (int p = 0; p < 2; ++p) {
    float bv = b2_2[p * 16 + ln15];
    v8f acc;
#pragma unroll
    for (int v = 0; v < 8; ++v) acc[v] = bv;
#pragma unroll
    for (int k = 0; k < HH; k += 4) {
      v2f a;
      a[0] = hW[ln15 * HH + k + kb];
      a[1] = hW[ln15 * HH + k + kb + 1];
      v2f b;
      b[0] = w2_2[(k + kb) * EE + p * 16 + ln15];
      b[1] = w2_2[(k + kb + 1) * EE + p * 16 + ln15];
      acc = wmma4(a, b, acc);
    }
#pragma unroll
    for (int v = 0; v < 8; ++v) w2f[p][v] = fabsf(acc[v]);
  }
  // b2 = relu(s @ wb2_1 + bb2_1) @ wb2_2 + bb2_2  (reduced in-wave)
  float t2[8];
#pragma unroll
  for (int v = 0; v < 8; ++v) t2[v] = 0.f;
  for (int p = 0; p < 2; ++p) {
    v8f acc = gemm_s(sW, wb2_1, EE, bb2_1, p * 16, ln15, kb);
    float wv = wb2_2[p * 16 + ln15];
#pragma unroll
    for (int v = 0; v < 8; ++v) t2[v] += fmaxf(acc[v], 0.f) * wv;
  }
#pragma unroll
  for (int v = 0; v < 8; ++v) {
    float s = t2[v];
    s += __shfl_xor(s, 1, 32);
    s += __shfl_xor(s, 2, 32);
    s += __shfl_xor(s, 4, 32);
    s += __shfl_xor(s, 8, 32);
    t2[v] = s + bb2_2[0];   // lanes 0-15: rows v ; lanes 16-31: rows v+8
  }
  // h1 = relu(s @ w1_1 + b1_1) -> lds_h (overwrites h2, same-wave DS order ok)
  for (int nt = 0; nt < 4; ++nt) {
    v8f acc = gemm_s(sW, w1_1, HH, b1_1, nt * 16, ln15, kb);
#pragma unroll
    for (int v = 0; v < 8; ++v)
      hW[(v + 8 * hi) * HH + nt * 16 + ln15] = fmaxf(acc[v], 0.f);
  }
  // hidden accumulator init = b1 = s @ wb1 + bb1 (fragment layout == hidden)
  v8f hacc[2];
  for (int p = 0; p < 2; ++p) hacc[p] = gemm_s(sW, wb1, EE, bb1, p * 16, ln15, kb);

  // preload the 16 A-fragments of h1 (16x64) into registers
  v2f af[16];
#pragma unroll
  for (int k = 0; k < 16; ++k) {
    af[k][0] = hW[ln15 * HH + k * 4 + kb];
    af[k][1] = hW[ln15 * HH + k * 4 + kb + 1];
  }

  // ---- Phase C: hidden += sum_n qs[:,n] * |h1 @ w1_2[:, n*32:(n+1)*32] + b1_2|
  for (int n = 0; n < NA; ++n) {
    __syncthreads();  // protect lds_w against previous iteration's readers
    for (int it = tid; it < HH * EE; it += 128)
      lds_w[it] = w1_2[(size_t)(it >> 5) * (NA * EE) + n * EE + (it & 31)];
    __syncthreads();
    float qv[8];
#pragma unroll
    for (int v = 0; v < 8; ++v)
      qv[v] = agent_qs[(size_t)(grow0 + v + 8 * hi) * NA + n];
#pragma unroll
    for (int p = 0; p < 2; ++p) {
      float bv = b1_2[n * EE + p * 16 + ln15];
      v8f d;
#pragma unroll
      for (int v = 0; v < 8; ++v) d[v] = bv;
#pragma unroll
      for (int k = 0; k < 16; ++k) {
        int kk = k * 4 + kb;
        v2f b;
        b[0] = lds_w[kk * EE + p * 16 + ln15];
        b[1] = lds_w[(kk + 1) * EE + p * 16 + ln15];
        d = wmma4(af[k], b, d);
      }
#pragma unroll
      for (int v = 0; v < 8; ++v) hacc[p][v] += fabsf(d[v]) * qv[v];
    }
  }

  // ---- Phase D: q_tot = sum_e elu(hidden) * w2 + b2 ----
  float t[8];
#pragma unroll
  for (int v = 0; v < 8; ++v) {
    float acc = 0.f;
#pragma unroll
    for (int p = 0; p < 2; ++p) {
      float x = hacc[p][v];
      float h = (x > 0.f) ? x : expm1f(x);
      acc += h * w2f[p][v];
    }
    acc += __shfl_xor(acc, 1, 32);
    acc += __shfl_xor(acc, 2, 32);
    acc += __shfl_xor(acc, 4, 32);
    acc += __shfl_xor(acc, 8, 32);
    t[v] = acc;
  }
  if (ln15 == 0) {   // lane 0 writes rows 0..7, lane 16 writes rows 8..15
#pragma unroll
    for (int v = 0; v < 8; ++v)
      out[grow0 + 8 * hi + v] = t[v] + t2[v];
  }
}

extern "C" void kernel_launch(void* const* d_in, const int* in_sizes, int n_in,
                              void* d_out, int out_size, void* d_ws, size_t ws_size,
                              hipStream_t stream) {
  (void)in_sizes; (void)n_in; (void)d_ws; (void)ws_size; (void)out_size;
  const float* agent_qs = (const float*)d_in[0];
  const float* obs      = (const float*)d_in[1];
  const int*   mask     = (const int*)  d_in[2];
  const float* w1_1  = (const float*)d_in[3];
  const float* b1_1  = (const float*)d_in[4];
  const float* w1_2  = (const float*)d_in[5];
  const float* b1_2  = (const float*)d_in[6];
  const float* w2_1  = (const float*)d_in[7];
  const float* b2_1  = (const float*)d_in[8];
  const float* w2_2  = (const float*)d_in[9];
  const float* b2_2  = (const float*)d_in[10];
  const float* wb1   = (const float*)d_in[11];
  const float* bb1   = (const float*)d_in[12];
  const float* wb2_1 = (const float*)d_in[13];
  const float* bb2_1 = (const float*)d_in[14];
  const float* wb2_2 = (const float*)d_in[15];
  const float* bb2_2 = (const float*)d_in[16];
  float* out = (float*)d_out;

  dim3 grid(BTOT / 64);   // 512 workgroups, each owns 64 rows
  dim3 block(128);        // 4 wave32s
  qmix_fused_kernel<<<grid, block, 0, stream>>>(
      agent_qs, obs, mask, w1_1, b1_1, w1_2, b1_2, w2_1, b2_1, w2_2, b2_2,
      wb1, bb1, wb2_1, bb2_1, wb2_2, bb2_2, out);
}